// LiteMLA_45930380263472
// MI455X (gfx1250) — compile-verified
//
#include <hip/hip_runtime.h>

// LiteMLA on gfx1250 (MI455X): fp32 WMMA 16x16x4 GEMMs with async
// global->LDS double-buffered staging (ASYNCcnt), fused depthwise+grouped
// pointwise, LDS-reduced linear attention.

typedef __attribute__((ext_vector_type(2))) float v2f;
typedef __attribute__((ext_vector_type(8))) float v8f;
typedef __attribute__((ext_vector_type(4))) int   v4i;

#define HW_N 4096   // 64*64

// ---------------------------------------------------------------------------
// Async global->LDS copy plumbing (gfx1250 GLOBAL_LOAD_ASYNC_TO_LDS_B128).
// Builtin signature: void(v4i addrspace(1)*, v4i addrspace(3)*, imm, imm).
// ---------------------------------------------------------------------------
#if __has_builtin(__builtin_amdgcn_global_load_async_to_lds_b128)
#define LITEMLA_ASYNC 1
#else
#define LITEMLA_ASYNC 0
#endif

#if LITEMLA_ASYNC
typedef __attribute__((address_space(1))) v4i as1_v4i;
typedef __attribute__((address_space(3))) v4i as3_v4i;
static __device__ __forceinline__ as1_v4i* as_global(const void* p) {
  // generic global pointers are numerically identical to AS1
  return (as1_v4i*)(unsigned long long)p;
}
static __device__ __forceinline__ as3_v4i* as_lds(const void* p) {
  // low 32 bits of a generic LDS address are the LDS byte offset
  return (as3_v4i*)(unsigned)(unsigned long long)p;
}
static __device__ __forceinline__ void async_copy16(const void* g, void* l) {
  __builtin_amdgcn_global_load_async_to_lds_b128(as_global(g), as_lds(l), 0, 0);
}
#endif

static __device__ __forceinline__ void wait_async_prev3() {
#if LITEMLA_ASYNC
#if __has_builtin(__builtin_amdgcn_s_wait_asynccnt)
  __builtin_amdgcn_s_wait_asynccnt(3);
#else
  asm volatile("s_wait_asynccnt 0x3" ::: "memory");
#endif
#endif
}
static __device__ __forceinline__ void wait_async_all() {
#if LITEMLA_ASYNC
#if __has_builtin(__builtin_amdgcn_s_wait_asynccnt)
  __builtin_amdgcn_s_wait_asynccnt(0);
#else
  asm volatile("s_wait_asynccnt 0x0" ::: "memory");
#endif
#endif
}

// ---------------------------------------------------------------------------
// GEMM: D[b][M][N] = A[M][K] * B[b][K][N]   (row-major, fp32)
// Block: 256 threads = 8 waves (4 M x 2 N). Block tile 128(M) x 64(N), BK=16.
// Each wave: 32x32 = 2x2 V_WMMA_F32_16X16X4_F32 accumulator tiles.
// Double-buffered LDS slabs filled by async B128 copies (3 per thread per
// slab -> s_wait_asynccnt 3 drains exactly the previous slab).
// LDS pitches: A row = 20 floats (80B), B row = 72 floats (288B): both 16B
// multiples (B128 alignment) and bank-conflict free for the fragment reads.
// ---------------------------------------------------------------------------
template <int M, int K>
__global__ __launch_bounds__(256) void gemm_wmma_f32(
    const float* __restrict__ A, const float* __restrict__ B,
    float* __restrict__ D, int N) {
  constexpr int BM = 128, BN = 64, BK = 16;
  constexpr int AP = BK + 4;   // 20
  constexpr int BP = BN + 8;   // 72
  __shared__ alignas(16) float As[2][BM][AP];
  __shared__ alignas(16) float Bs[2][BK][BP];

  const int t     = threadIdx.x;
  const int lane  = t & 31;
  const int wave  = t >> 5;
  const int half  = lane >> 4;   // 0 / 1
  const int lm    = lane & 15;
  const int waveM = wave >> 1;   // 0..3
  const int waveN = wave & 1;    // 0..1

  const int m0 = blockIdx.y * BM;
  const int n0 = blockIdx.x * BN;
  const int b  = blockIdx.z;
  const float* Bb = B + (size_t)b * K * N;
  float*       Db = D + (size_t)b * (size_t)M * N;

  v8f acc[2][2];
#pragma unroll
  for (int i = 0; i < 2; ++i)
#pragma unroll
    for (int j = 0; j < 2; ++j) acc[i][j] = (v8f)0.f;

  const int ar = t >> 2;         // 0..63  (A tile row, x2 passes)
  const int ac = (t & 3) * 4;    // 0,4,8,12
  const int br = t >> 4;         // 0..15  (B tile row)
  const int bc = (t & 15) * 4;   // 0..60

  auto stage = [&](int tile, int buf) {
    const float* ga0 = A + (size_t)(m0 + ar) * K + tile * BK + ac;
    const float* ga1 = ga0 + (size_t)64 * K;
    const float* gb  = Bb + (size_t)(tile * BK + br) * N + n0 + bc;
#if LITEMLA_ASYNC
    async_copy16(ga0, &As[buf][ar][ac]);
    async_copy16(ga1, &As[buf][ar + 64][ac]);
    async_copy16(gb,  &Bs[buf][br][bc]);
#else
    *(float4*)&As[buf][ar][ac]      = *(const float4*)ga0;
    *(float4*)&As[buf][ar + 64][ac] = *(const float4*)ga1;
    *(float4*)&Bs[buf][br][bc]      = *(const float4*)gb;
#endif
  };

  constexpr int NT = K / BK;
  stage(0, 0);

  for (int tI = 0; tI < NT; ++tI) {
    const int cur = tI & 1;
    if (tI + 1 < NT) {
      stage(tI + 1, cur ^ 1);   // overlap next slab copy with this compute
      wait_async_prev3();       // previous slab's 3 copies complete
    } else {
      wait_async_all();
    }
    __syncthreads();

#pragma unroll
    for (int kk = 0; kk < BK; kk += 4) {
      v2f afr[2], bfr[2];
#pragma unroll
      for (int mi = 0; mi < 2; ++mi) {
        const int mrow = waveM * 32 + mi * 16 + lm;
        afr[mi].x = As[cur][mrow][kk + 2 * half];
        afr[mi].y = As[cur][mrow][kk + 2 * half + 1];
      }
#pragma unroll
      for (int ni = 0; ni < 2; ++ni) {
        const int ncol = waveN * 32 + ni * 16 + lm;
        bfr[ni].x = Bs[cur][kk + 2 * half][ncol];
        bfr[ni].y = Bs[cur][kk + 2 * half + 1][ncol];
      }
#pragma unroll
      for (int mi = 0; mi < 2; ++mi)
#pragma unroll
        for (int ni = 0; ni < 2; ++ni)
          acc[mi][ni] = __builtin_amdgcn_wmma_f32_16x16x4_f32(
              false, afr[mi], false, bfr[ni], (short)0, acc[mi][ni],
              false, false);
    }
    __syncthreads();  // all waves done reading buf `cur` before it refills
  }

  // ---- store D tiles (coalesced: lm -> consecutive columns) ----
#pragma unroll
  for (int mi = 0; mi < 2; ++mi)
#pragma unroll
    for (int ni = 0; ni < 2; ++ni) {
      const int tm = m0 + waveM * 32 + mi * 16;
      const int tn = n0 + waveN * 32 + ni * 16 + lm;
#pragma unroll
      for (int r = 0; r < 8; ++r) {
        const int row = tm + half * 8 + r;
        Db[(size_t)row * N + tn] = acc[mi][ni][r];
      }
    }
}

// ---------------------------------------------------------------------------
// Fused depthwise 5x5 (pad 2) + grouped 1x1 (8->8 per group, 96 groups).
// Block = one (batch, group, 16x16 pixel tile); depthwise results in LDS.
// ---------------------------------------------------------------------------
__global__ __launch_bounds__(256) void dw_pw_kernel(
    const float* __restrict__ qkv, const float* __restrict__ w_dw,
    const float* __restrict__ w_pw, float* __restrict__ agg) {
  __shared__ float dwv[8][256];
  const int t    = threadIdx.x;
  const int tile = blockIdx.x;   // 0..15 (4x4 tiles of 16x16)
  const int g    = blockIdx.y;   // 0..95
  const int b    = blockIdx.z;
  const int th   = (tile >> 2) * 16 + (t >> 4);
  const int tw   = (tile & 3) * 16 + (t & 15);
  const int pix  = th * 64 + tw;

  const float* src = qkv + ((size_t)b * 768 + g * 8) * HW_N;
#pragma unroll
  for (int ic = 0; ic < 8; ++ic) {
    const float* sp = src + (size_t)ic * HW_N;
    const float* wd = w_dw + (g * 8 + ic) * 25;
    float a = 0.f;
#pragma unroll
    for (int dy = 0; dy < 5; ++dy) {
      const int yy = th + dy - 2;
      if (yy < 0 || yy > 63) continue;
#pragma unroll
      for (int dx = 0; dx < 5; ++dx) {
        const int xx = tw + dx - 2;
        if (xx < 0 || xx > 63) continue;
        a = fmaf(sp[yy * 64 + xx], wd[dy * 5 + dx], a);
      }
    }
    dwv[ic][t] = a;
  }
  __syncthreads();

  float* dst = agg + ((size_t)b * 768 + g * 8) * HW_N;
#pragma unroll
  for (int oc = 0; oc < 8; ++oc) {
    const float* wp = w_pw + (g * 8 + oc) * 8;
    float s = 0.f;
#pragma unroll
    for (int ic = 0; ic < 8; ++ic) s = fmaf(wp[ic], dwv[ic][t], s);
    dst[(size_t)oc * HW_N + pix] = s;
  }
}

// ---------------------------------------------------------------------------
// Linear attention per (b, g): kv = relu(k)^T [v|1] over N=4096, then
// out = relu(q) @ kv, out[:, :8] / (out[:, 8] + eps).
// ms group g<32 comes from qkv channels g*24.., g>=32 from agg (g-32)*24..
// ---------------------------------------------------------------------------
__global__ __launch_bounds__(256) void linatt_kernel(
    const float* __restrict__ qkv, const float* __restrict__ agg,
    float* __restrict__ outb) {
  const int t = threadIdx.x;
  const int g = blockIdx.x;   // 0..63
  const int b = blockIdx.y;   // 0..7
  const float* base = (g < 32)
      ? (qkv + ((size_t)b * 768 + g * 24) * HW_N)
      : (agg + ((size_t)b * 768 + (g - 32) * 24) * HW_N);
  const float* Q  = base;
  const float* Kp = base + (size_t)8 * HW_N;
  const float* V  = base + (size_t)16 * HW_N;

  __shared__ float part[72 * 257];  // stride 257 -> conflict-free reduce
  __shared__ float kvf[72];

  float acc[72];
#pragma unroll
  for (int p = 0; p < 72; ++p) acc[p] = 0.f;

  // Phase 1: per-thread 8x9 outer-product accumulation over n = t + 256*i
  for (int i = 0; i < 16; ++i) {
    const int n = t + i * 256;
    float kd[8], vv[9];
#pragma unroll
    for (int d = 0; d < 8; ++d) {
      const float kk = Kp[(size_t)d * HW_N + n];
      kd[d] = kk > 0.f ? kk : 0.f;
    }
#pragma unroll
    for (int e = 0; e < 8; ++e) vv[e] = V[(size_t)e * HW_N + n];
    vv[8] = 1.f;
#pragma unroll
    for (int d = 0; d < 8; ++d)
#pragma unroll
      for (int e = 0; e < 9; ++e)
        acc[d * 9 + e] = fmaf(kd[d], vv[e], acc[d * 9 + e]);
  }
#pragma unroll
  for (int p = 0; p < 72; ++p) part[p * 257 + t] = acc[p];
  __syncthreads();
  if (t < 72) {
    float s = 0.f;
    for (int i = 0; i < 256; ++i) s += part[t * 257 + i];
    kvf[t] = s;
  }
  __syncthreads();

  // Phase 2: out[n, e] = relu(q) . kv, normalized by the appended row
  float* dst = outb + ((size_t)b * 64 + g) * 8 * HW_N;
  for (int i = 0; i < 16; ++i) {
    const int n = t + i * 256;
    float qd[8];
#pragma unroll
    for (int d = 0; d < 8; ++d) {
      const float qq = Q[(size_t)d * HW_N + n];
      qd[d] = qq > 0.f ? qq : 0.f;
    }
    float denom = 0.f;
#pragma unroll
    for (int d = 0; d < 8; ++d) denom = fmaf(qd[d], kvf[d * 9 + 8], denom);
    const float inv = 1.f / (denom + 1e-15f);
#pragma unroll
    for (int e = 0; e < 8; ++e) {
      float s = 0.f;
#pragma unroll
      for (int d = 0; d < 8; ++d) s = fmaf(qd[d], kvf[d * 9 + e], s);
      dst[(size_t)e * HW_N + n] = s * inv;
    }
  }
}

// ---------------------------------------------------------------------------
extern "C" void kernel_launch(void* const* d_in, const int* in_sizes, int n_in,
                              void* d_out, int out_size, void* d_ws,
                              size_t ws_size, hipStream_t stream) {
  const float* x      = (const float*)d_in[0];   // [8,256,64,64]
  const float* w_qkv  = (const float*)d_in[1];   // [768,256]
  const float* w_dw   = (const float*)d_in[2];   // [768,25]
  const float* w_pw   = (const float*)d_in[3];   // [768,8]
  const float* w_proj = (const float*)d_in[4];   // [256,512]
  float* out = (float*)d_out;                    // [8,256,64,64]

  float* qkv  = (float*)d_ws;                          // 8*768*4096 f32
  float* agg  = qkv  + (size_t)8 * 768 * HW_N;         // 8*768*4096 f32
  float* attn = agg  + (size_t)8 * 768 * HW_N;         // 8*512*4096 f32

  const int N = HW_N;
  dim3 blk(256);

  // qkv = W_qkv[768x256] @ x[b][256x4096]
  gemm_wmma_f32<768, 256>
      <<<dim3(N / 64, 768 / 128, 8), blk, 0, stream>>>(w_qkv, x, qkv, N);
  // agg = grouped_pw(depthwise5x5(qkv))
  dw_pw_kernel<<<dim3(16, 96, 8), blk, 0, stream>>>(qkv, w_dw, w_pw, agg);
  // linear attention over 64 head-groups per batch
  linatt_kernel<<<dim3(64, 8), blk, 0, stream>>>(qkv, agg, attn);
  // out = W_proj[256x512] @ attn[b][512x4096]
  gemm_wmma_f32<256, 512>
      <<<dim3(N / 64, 256 / 128, 8), blk, 0, stream>>>(w_proj, attn, out, N);
}